// ChamferDistance_17952963297894
// MI455X (gfx1250) — compile-verified
//
#include <hip/hip_runtime.h>
#include <math.h>

typedef __attribute__((ext_vector_type(2))) float v2f;
typedef __attribute__((ext_vector_type(8))) float v8f;

#define BATCH 4
#define NPTS  4096
#define NT    256            // 16-point tiles per cloud
#define WAVES_PER_BLOCK 4
#define TPB   (WAVES_PER_BLOCK * 32)

// One directed Chamfer pass: for every point m of P, min over all n of
// ||P_m - Q_n||, written (already sqrt'ed) to outMin[batch*NPTS + m].
//
// Trick: with A row m = (x,y,z,1), B col n = (x',y',z',-|q|^2/2) and
// C[m][n] = -|p|^2/2, a single V_WMMA_F32_16X16X4_F32 yields
//   D[m][n] = p.q - |q|^2/2 - |p|^2/2 = -dist^2/2,
// so min dist^2 = -2 * max D.  Inner loop: 1 ds_load_b64 + 1 wmma + 8 max.
__global__ __launch_bounds__(TPB) void chamfer_dir_kernel(
    const float* __restrict__ P,
    const float* __restrict__ Q,
    float* __restrict__ outMin)
{
    // B operand pre-swizzled into WMMA lane layout: for tile j, lane l reads
    // sB[j*32 + l].  Lanes 0-15: (x,y) = (K0,K1); lanes 16-31: (z,-sq/2) = (K2,K3).
    __shared__ float2 sB[NT * 32];   // exactly 64 KB

    const int batch = blockIdx.x / (NT / WAVES_PER_BLOCK);
    const int tg    = blockIdx.x % (NT / WAVES_PER_BLOCK);
    const int tid   = threadIdx.x;
    const int wave  = tid >> 5;
    const int lane  = tid & 31;

    const float* Pb = P + (size_t)batch * NPTS * 3;
    const float* Qb = Q + (size_t)batch * NPTS * 3;

    // Stage opposing cloud into LDS, already in B-operand layout.
    for (int p = tid; p < NPTS; p += TPB) {
        const float x  = Qb[3 * p + 0];
        const float y  = Qb[3 * p + 1];
        const float z  = Qb[3 * p + 2];
        const float sq = fmaf(x, x, fmaf(y, y, z * z));
        const int j = p >> 4;
        const int c = p & 15;
        sB[j * 32 + c]      = make_float2(x, y);
        sB[j * 32 + 16 + c] = make_float2(z, -0.5f * sq);
    }
    __syncthreads();

    // This wave's 16-row i-tile (A operand), loop-invariant.
    const int itile = tg * WAVES_PER_BLOCK + wave;
    const int i0    = itile * 16;
    const int m     = i0 + (lane & 15);
    const float px = Pb[3 * m + 0];
    const float py = Pb[3 * m + 1];
    const float pz = Pb[3 * m + 2];
    const float sp = fmaf(px, px, fmaf(py, py, pz * pz));

    const bool lo = (lane < 16);
    // A-matrix 16x4 layout: lanes 0-15 hold (K0,K1)=(x,y); lanes 16-31 (K2,K3)=(z,1).
    v2f a;
    a.x = lo ? px : pz;
    a.y = lo ? py : 1.0f;

    // C operand: C[v][lane] = -|p|^2/2 for M = v (+8 in the upper half).
    // C/D VGPR v holds M=v (lanes 0-15) or M=v+8 (lanes 16-31).
    const int hofs = (lane & 16) >> 1;   // 0 or 8
    v8f cacc;
    #pragma unroll
    for (int v = 0; v < 8; ++v)
        cacc[v] = -0.5f * __shfl(sp, v + hofs, 32);

    float runmax[8];
    #pragma unroll
    for (int v = 0; v < 8; ++v) runmax[v] = -3.0e38f;

    const float2* bp = sB + lane;
    #pragma unroll 8
    for (int j = 0; j < NT; ++j) {
        const float2 qb = bp[j * 32];        // one ds_load_b64, conflict-free
        v2f b;
        b.x = qb.x;
        b.y = qb.y;
        // (neg_a, A, neg_b, B, c_mod, C, reuse_a, reuse_b)
        const v8f d = __builtin_amdgcn_wmma_f32_16x16x4_f32(
            false, a, false, b, (short)0, cacc, false, false);
        #pragma unroll
        for (int v = 0; v < 8; ++v)
            runmax[v] = fmaxf(runmax[v], d[v]);
    }

    // Reduce across the 16 lanes of each half (N dimension); xor masks <=8 stay
    // inside a 16-lane group on wave32.  min dist^2 = -2 * max D.
    #pragma unroll
    for (int v = 0; v < 8; ++v) {
        float r = runmax[v];
        #pragma unroll
        for (int off = 8; off >= 1; off >>= 1)
            r = fmaxf(r, __shfl_xor(r, off, 32));
        const float dmin = sqrtf(fmaxf(-2.0f * r, 0.0f));
        if (lane == 0)        outMin[batch * NPTS + i0 + v]     = dmin;
        else if (lane == 16)  outMin[batch * NPTS + i0 + v + 8] = dmin;
    }
}

// Deterministic single-block sum of all 2*B*N per-point minima.
__global__ __launch_bounds__(256) void chamfer_reduce_kernel(
    const float* __restrict__ mins, float* __restrict__ out)
{
    __shared__ float red[256];
    const int t = threadIdx.x;
    const int M = 2 * BATCH * NPTS;
    float s = 0.0f;
    for (int i = t; i < M; i += 256) s += mins[i];
    red[t] = s;
    __syncthreads();
    for (int w = 128; w > 0; w >>= 1) {
        if (t < w) red[t] += red[t + w];
        __syncthreads();
    }
    if (t == 0)
        out[0] = red[0] / (2.0f * (float)NPTS * (float)BATCH);
}

extern "C" void kernel_launch(void* const* d_in, const int* in_sizes, int n_in,
                              void* d_out, int out_size, void* d_ws, size_t ws_size,
                              hipStream_t stream)
{
    const float* pc1 = (const float*)d_in[0];
    const float* pc2 = (const float*)d_in[1];
    float* mins = (float*)d_ws;                 // 2*B*N floats = 128 KB scratch

    dim3 grid(BATCH * (NT / WAVES_PER_BLOCK));  // 256 blocks, 4 waves each
    chamfer_dir_kernel<<<grid, TPB, 0, stream>>>(pc1, pc2, mins);
    chamfer_dir_kernel<<<grid, TPB, 0, stream>>>(pc2, pc1, mins + BATCH * NPTS);

    chamfer_reduce_kernel<<<1, 256, 0, stream>>>(mins, (float*)d_out);
}